// SwavBase_32615981646586
// MI455X (gfx1250) — compile-verified
//
#include <hip/hip_runtime.h>
#include <hip/hip_bf16.h>

typedef float v2f __attribute__((ext_vector_type(2)));
typedef float v8f __attribute__((ext_vector_type(8)));

constexpr int NCELL  = 65536;   // n_cells
constexpr int DIM    = 128;     // latent dim (reduction)
constexpr int KPROTO = 2048;    // prototypes
constexpr int MT = 64;          // rows per block tile
constexpr int NT = 64;          // cols per block tile
constexpr unsigned INF_BITS = 0x7F800000u;

// ---------------------------------------------------------------------------
// K0: init row/col min arrays (float-as-uint ordered trick, dist >= 0)
// ---------------------------------------------------------------------------
__global__ void k_init_mins(unsigned* __restrict__ mins, int n) {
    int i = blockIdx.x * blockDim.x + threadIdx.x;
    if (i < n) mins[i] = INF_BITS;
}

// ---------------------------------------------------------------------------
// K1: one wave32 per row: z = x / max(||x||,1e-12); zz = sum(z*z) (post-round)
// ---------------------------------------------------------------------------
__global__ void k_rownorm(const float* __restrict__ x, float* __restrict__ z,
                          float* __restrict__ zz) {
    const int row  = (blockIdx.x * blockDim.x + threadIdx.x) >> 5;
    const int lane = threadIdx.x & 31;
    const float4 xv = reinterpret_cast<const float4*>(x)[(size_t)row * 32 + lane];
    float s = xv.x * xv.x + xv.y * xv.y + xv.z * xv.z + xv.w * xv.w;
#pragma unroll
    for (int off = 16; off > 0; off >>= 1) s += __shfl_xor(s, off, 32);
    const float scale = 1.0f / fmaxf(sqrtf(s), 1e-12f);
    const float4 zv = make_float4(xv.x * scale, xv.y * scale, xv.z * scale, xv.w * scale);
    reinterpret_cast<float4*>(z)[(size_t)row * 32 + lane] = zv;
    float s2 = zv.x * zv.x + zv.y * zv.y + zv.z * zv.z + zv.w * zv.w;
#pragma unroll
    for (int off = 16; off > 0; off >>= 1) s2 += __shfl_xor(s2, off, 32);
    if (lane == 0) zz[row] = s2;
}

// ---------------------------------------------------------------------------
// K2: one wave32 per prototype row: ww = sum(W*W)
// ---------------------------------------------------------------------------
__global__ void k_wnorm(const float* __restrict__ W, float* __restrict__ ww) {
    const int row  = (blockIdx.x * blockDim.x + threadIdx.x) >> 5;
    const int lane = threadIdx.x & 31;
    const float4 wv = reinterpret_cast<const float4*>(W)[(size_t)row * 32 + lane];
    float s = wv.x * wv.x + wv.y * wv.y + wv.z * wv.z + wv.w * wv.w;
#pragma unroll
    for (int off = 16; off > 0; off >>= 1) s += __shfl_xor(s, off, 32);
    if (lane == 0) ww[row] = s;
}

// ---------------------------------------------------------------------------
// K3: WMMA f32 GEMM logits = z @ W^T, fused sqrt-distance + row/col min.
//     128 threads (4 waves), 64x64 tile, full D=128 staged in LDS.
// ---------------------------------------------------------------------------
__global__ __launch_bounds__(128)
void k_gemm_dist(const float* __restrict__ z, const float* __restrict__ W,
                 const float* __restrict__ zz, const float* __restrict__ ww,
                 float* __restrict__ logits,
                 unsigned* __restrict__ rowmin, unsigned* __restrict__ colmin) {
    __shared__ float As[MT][DIM + 4];   // stride 132: conflict-free fragment reads
    __shared__ float Bs[NT][DIM + 4];   // Bs[n][k] = W[colBase+n][k]
    __shared__ float zz_s[MT];
    __shared__ float ww_s[NT];
    __shared__ unsigned cmin_s[NT];

    const int tid      = threadIdx.x;
    const int lane     = tid & 31;
    const int wv       = tid >> 5;      // wave 0..3
    const int laneHalf = lane >> 4;     // 0|1
    const int nIdx     = lane & 15;
    const int rowBase  = blockIdx.x * MT;
    const int colBase  = blockIdx.y * NT;

    // cooperative tile load: 64 rows x 32 float4 per array, 16 float4/thread
    for (int i = tid; i < MT * (DIM / 4); i += 128) {
        const int r  = i >> 5;          // 0..63
        const int kq = i & 31;          // float4 index along D
        const float4 av = reinterpret_cast<const float4*>(z + (size_t)(rowBase + r) * DIM)[kq];
        *reinterpret_cast<float4*>(&As[r][kq << 2]) = av;
        const float4 bv = reinterpret_cast<const float4*>(W + (size_t)(colBase + r) * DIM)[kq];
        *reinterpret_cast<float4*>(&Bs[r][kq << 2]) = bv;
    }
    if (tid < NT) {
        ww_s[tid]   = ww[colBase + tid];
        cmin_s[tid] = INF_BITS;
    } else if (tid < NT + MT) {
        zz_s[tid - NT] = zz[rowBase + (tid - NT)];
    }
    __syncthreads();

    // --- WMMA main loop: V_WMMA_F32_16X16X4_F32, 32 k-steps ---
    // A frag (16x4 f32): lane m=lane%16 holds row m; v0=K(2h), v1=K(2h+1)
    // B frag (4x16 f32): lane n=lane%16 holds col n; v0=K(2h), v1=K(2h+1)
    v8f acc[4] = {};
#pragma unroll 4
    for (int k0 = 0; k0 < DIM; k0 += 4) {
        const int ka = k0 + (laneHalf << 1);
        const v2f a = *reinterpret_cast<const v2f*>(&As[(wv << 4) + nIdx][ka]);
#pragma unroll
        for (int c = 0; c < 4; ++c) {
            const v2f b = *reinterpret_cast<const v2f*>(&Bs[(c << 4) + nIdx][ka]);
            acc[c] = __builtin_amdgcn_wmma_f32_16x16x4_f32(
                false, a, false, b, (short)0, acc[c], false, false);
        }
    }

    // --- epilogue: NT store of logits + fused distance & min reductions ---
    // C/D layout: VGPR r -> (M = r + 8*laneHalf, N = nIdx)
    const float INF = __builtin_inff();
    float cmin[4] = {INF, INF, INF, INF};
#pragma unroll
    for (int r = 0; r < 8; ++r) {
        const int rl  = (wv << 4) + r + (laneHalf << 3);   // local row 0..63
        const int row = rowBase + rl;
        const float zzr = zz_s[rl];
        float rmin = INF;
#pragma unroll
        for (int c = 0; c < 4; ++c) {
            const int cl = (c << 4) + nIdx;                // local col 0..63
            const float v = acc[c][r];
            __builtin_nontemporal_store(v, &logits[(size_t)row * KPROTO + colBase + cl]);
            const float d = sqrtf(fmaxf(zzr + ww_s[cl] - 2.0f * v, 0.0f));
            rmin    = fminf(rmin, d);
            cmin[c] = fminf(cmin[c], d);
        }
        // row-min across the 16 lanes of this half (xor 1,2,4,8 stays in-half)
        rmin = fminf(rmin, __shfl_xor(rmin, 1, 32));
        rmin = fminf(rmin, __shfl_xor(rmin, 2, 32));
        rmin = fminf(rmin, __shfl_xor(rmin, 4, 32));
        rmin = fminf(rmin, __shfl_xor(rmin, 8, 32));
        if (nIdx == 0) atomicMin(&rowmin[row], __float_as_uint(rmin));
    }
    // col-min: combine halves (different rows, same cols), stage in LDS
#pragma unroll
    for (int c = 0; c < 4; ++c) {
        const float m = fminf(cmin[c], __shfl_xor(cmin[c], 16, 32));
        if (laneHalf == 0) atomicMin(&cmin_s[(c << 4) + nIdx], __float_as_uint(m));
    }
    __syncthreads();
    if (tid < NT) atomicMin(&colmin[colBase + tid], cmin_s[tid]);
}

// ---------------------------------------------------------------------------
// K4: means of min arrays + cvae scalar
// ---------------------------------------------------------------------------
__global__ void k_finalize(const unsigned* __restrict__ rowmin,
                           const unsigned* __restrict__ colmin,
                           const float* __restrict__ recon, const float* __restrict__ kl,
                           const float* __restrict__ mmd, float* __restrict__ outs) {
    __shared__ float red[256];
    float s = 0.0f;
    for (int i = threadIdx.x; i < NCELL; i += 256) s += __uint_as_float(rowmin[i]);
    red[threadIdx.x] = s;
    __syncthreads();
    for (int st = 128; st > 0; st >>= 1) {
        if ((int)threadIdx.x < st) red[threadIdx.x] += red[threadIdx.x + st];
        __syncthreads();
    }
    const float prop = red[0] * (1.0f / NCELL);
    __syncthreads();
    s = 0.0f;
    for (int i = threadIdx.x; i < KPROTO; i += 256) s += __uint_as_float(colmin[i]);
    red[threadIdx.x] = s;
    __syncthreads();
    for (int st = 128; st > 0; st >>= 1) {
        if ((int)threadIdx.x < st) red[threadIdx.x] += red[threadIdx.x + st];
        __syncthreads();
    }
    if (threadIdx.x == 0) {
        outs[0] = recon[0] + 0.5f * kl[0] + mmd[0];   // cvae_loss
        outs[1] = prop;                                // propagation
        outs[2] = red[0] * (1.0f / KPROTO);            // emb_similarity
    }
}

// ---------------------------------------------------------------------------
extern "C" void kernel_launch(void* const* d_in, const int* in_sizes, int n_in,
                              void* d_out, int out_size, void* d_ws, size_t ws_size,
                              hipStream_t stream) {
    const float* x     = (const float*)d_in[0];
    const float* Wp    = (const float*)d_in[1];
    const float* recon = (const float*)d_in[2];
    const float* kl    = (const float*)d_in[3];
    const float* mmd   = (const float*)d_in[4];

    float* out     = (float*)d_out;
    float* z       = out;                                   // [N, D]
    float* logits  = out + (size_t)NCELL * DIM;             // [N, K]
    float* scalars = logits + (size_t)NCELL * KPROTO;       // cvae, prop, emb

    unsigned* rowmin = (unsigned*)d_ws;            // [NCELL]
    unsigned* colmin = rowmin + NCELL;             // [KPROTO]
    float*    zz     = (float*)(colmin + KPROTO);  // [NCELL]
    float*    ww     = zz + NCELL;                 // [KPROTO]

    k_init_mins<<<(NCELL + KPROTO + 255) / 256, 256, 0, stream>>>(rowmin, NCELL + KPROTO);
    k_rownorm<<<NCELL / 8, 256, 0, stream>>>(x, z, zz);
    k_wnorm<<<KPROTO / 8, 256, 0, stream>>>(Wp, ww);

    dim3 grid(NCELL / MT, KPROTO / NT);   // (1024, 32)
    k_gemm_dist<<<grid, 128, 0, stream>>>(z, Wp, zz, ww, logits, rowmin, colmin);

    k_finalize<<<1, 256, 0, stream>>>(rowmin, colmin, recon, kl, mmd, scalars);
}